// PPEG_80573586473418
// MI455X (gfx1250) — compile-verified
//
#include <hip/hip_runtime.h>
#include <cstdint>

// PPEG depthwise conv fusion for MI455X (gfx1250, wave32).
//
// Roofline: 5.6 GFLOP fp32 vs 1.07 GB of mandatory fp32 I/O -> AI ~5.2 F/B,
// memory-bound with a ~46 us HBM floor at 23.3 TB/s. The MI455X-specific win
// is the data path: stage halo tiles into the 320KB-per-WGP LDS with the
// gfx1250 async copy (global_load_async_to_lds_b128, ASYNCcnt), keep loads
// channel-contiguous (coalesced), and do register-blocked fp32 FMAs from LDS.
// WMMA is deliberately NOT used: depthwise conv has per-channel weights AND
// per-channel data, so no contraction dim is shared across a 16-wide tile;
// a Toeplitz-band WMMA mapping would force stride-2KB per-channel gathers.

namespace {
constexpr int Hh = 64, Ww = 64, Cc = 512, Bb = 16;
constexpr int Nn = 1 + Hh * Ww;          // 4097 tokens
constexpr int CG = 32;                   // channels per block (== wave width)
constexpr int NCG = Cc / CG;             // 16 channel groups
constexpr int TILE_R = 7;                // input rows (h-3 .. h+3)
constexpr int TILE_Q = 70;               // input cols (w-3 .. w+66), padded
constexpr int TILE_F = TILE_R * TILE_Q * CG;   // 15680 floats = 61.25 KB
constexpr int TILE_V4 = TILE_F / 4;            // 3920 float4
}

__global__ __launch_bounds__(256) void ppeg_dwconv_kernel(
    const float* __restrict__ x,
    const float* __restrict__ w7, const float* __restrict__ b7,
    const float* __restrict__ w5, const float* __restrict__ b5,
    const float* __restrict__ w3, const float* __restrict__ b3,
    float* __restrict__ out)
{
    __shared__ __align__(16) float smem[TILE_F];   // [r][q][cl]

    const int tid = threadIdx.x;
    const int cg  = blockIdx.x;          // channel group 0..15
    const int h   = blockIdx.y;          // output row 0..63
    const int b   = blockIdx.z;          // batch 0..15
    const int cl  = tid & 31;            // channel within group (lane id)
    const int w0  = (tid >> 5) * 8;      // 8 w-outputs per thread (8 waves)
    const int c   = cg * CG + cl;        // absolute channel

    // ---- 1) single-pass tile fill: async-load in-range, ds-zero the halo.
    // The two target sets are disjoint, so no pre-zero pass / extra barrier.
    float4* s4 = (float4*)smem;
    float4 z; z.x = z.y = z.z = z.w = 0.0f;
    const unsigned lds_base = (unsigned)(unsigned long long)(uintptr_t)&smem[0];
    for (int j = tid; j < TILE_V4; j += 256) {
        const int c4 = j & 7;                    // which float4 of the 32 ch
        const int q  = (j >> 3) % TILE_Q;        // padded col 0..69
        const int r  = j / (TILE_Q * 8);         // tile row 0..6
        const int hr = h - 3 + r;                // global image row
        const int wq = q - 3;                    // global image col
        if ((unsigned)hr < 64u && (unsigned)wq < 64u) {
            const float* gp = x + ((size_t)b * Nn + 1 + (size_t)hr * Ww + wq) * Cc
                                + (size_t)cg * CG + c4 * 4;
            const unsigned ldsoff = lds_base + (unsigned)(j * 16);
            const unsigned long long ga = (unsigned long long)(uintptr_t)gp;
            asm volatile("global_load_async_to_lds_b128 %0, %1, off"
                         :: "v"(ldsoff), "v"(ga) : "memory");
        } else {
            s4[j] = z;                           // zero padding (halo only)
        }
    }
#if __has_builtin(__builtin_amdgcn_s_wait_asynccnt)
    __builtin_amdgcn_s_wait_asynccnt(0);
#else
    asm volatile("s_wait_asynccnt 0" ::: "memory");
#endif
    __syncthreads();

    // ---- 2) per-channel filter taps into registers (170 KB total, L2-hot)
    float wr7[49], wr5[25], wr3[9];
#pragma unroll
    for (int k = 0; k < 49; ++k) wr7[k] = w7[(size_t)c * 49 + k];
#pragma unroll
    for (int k = 0; k < 25; ++k) wr5[k] = w5[(size_t)c * 25 + k];
#pragma unroll
    for (int k = 0; k < 9;  ++k) wr3[k] = w3[(size_t)c * 9 + k];
    const float bias = b7[c] + b5[c] + b3[c];

    float acc[8];
#pragma unroll
    for (int i = 0; i < 8; ++i) acc[i] = bias;

    // ---- 3) register-blocked fused 7/5/3/identity sweep over 7 rows ------
    // Per thread: 98 bank-conflict-free ds_loads feeding ~670 v_fma.
#pragma unroll
    for (int kh = 0; kh < 7; ++kh) {
        float v[14];                                  // cols w0 .. w0+13
#pragma unroll
        for (int j = 0; j < 14; ++j)
            v[j] = smem[(kh * TILE_Q + w0 + j) * CG + cl];

        // 7x7: output i uses cols i..i+6
#pragma unroll
        for (int i = 0; i < 8; ++i)
#pragma unroll
            for (int kw = 0; kw < 7; ++kw)
                acc[i] = fmaf(wr7[kh * 7 + kw], v[i + kw], acc[i]);

        // 5x5 lives on tile rows 1..5, col offset +1
        if (kh >= 1 && kh <= 5) {
            const int r5 = kh - 1;
#pragma unroll
            for (int i = 0; i < 8; ++i)
#pragma unroll
                for (int kw = 0; kw < 5; ++kw)
                    acc[i] = fmaf(wr5[r5 * 5 + kw], v[i + kw + 1], acc[i]);
        }
        // 3x3 lives on tile rows 2..4, col offset +2
        if (kh >= 2 && kh <= 4) {
            const int r3 = kh - 2;
#pragma unroll
            for (int i = 0; i < 8; ++i)
#pragma unroll
                for (int kw = 0; kw < 3; ++kw)
                    acc[i] = fmaf(wr3[r3 * 3 + kw], v[i + kw + 2], acc[i]);
        }
        // identity term: center row, center col
        if (kh == 3) {
#pragma unroll
            for (int i = 0; i < 8; ++i) acc[i] += v[i + 3];
        }
    }

    // ---- 4) coalesced store (lanes = consecutive channels) ---------------
#pragma unroll
    for (int i = 0; i < 8; ++i) {
        const size_t o = ((size_t)b * Nn + 1 + (size_t)(h * Ww + w0 + i)) * Cc + c;
        out[o] = acc[i];
    }
}

__global__ void ppeg_cls_copy_kernel(const float* __restrict__ x,
                                     float* __restrict__ out)
{
    const int t = blockIdx.x * blockDim.x + threadIdx.x;
    if (t < Bb * Cc) {
        const int bb = t / Cc, cch = t % Cc;
        const size_t off = (size_t)bb * Nn * Cc + cch;   // token 0 of batch bb
        out[off] = x[off];
    }
}

extern "C" void kernel_launch(void* const* d_in, const int* in_sizes, int n_in,
                              void* d_out, int out_size, void* d_ws, size_t ws_size,
                              hipStream_t stream)
{
    (void)in_sizes; (void)n_in; (void)out_size; (void)d_ws; (void)ws_size;
    const float* x  = (const float*)d_in[0];
    const float* w7 = (const float*)d_in[1];
    const float* b7 = (const float*)d_in[2];
    const float* w5 = (const float*)d_in[3];
    const float* b5 = (const float*)d_in[4];
    const float* w3 = (const float*)d_in[5];
    const float* b3 = (const float*)d_in[6];
    float* out = (float*)d_out;

    dim3 grid(NCG, Hh, Bb);              // 16 x 64 x 16 = 16384 workgroups
    ppeg_dwconv_kernel<<<grid, 256, 0, stream>>>(x, w7, b7, w5, b5, w3, b3, out);
    ppeg_cls_copy_kernel<<<(Bb * Cc + 255) / 256, 256, 0, stream>>>(x, out);
}